// KCenter_15642270892469
// MI455X (gfx1250) — compile-verified
//
#include <hip/hip_runtime.h>
#include <hip/hip_bf16.h>

typedef __attribute__((ext_vector_type(2))) float v2f;
typedef __attribute__((ext_vector_type(4))) float v4f;
typedef __attribute__((ext_vector_type(8))) float v8f;
typedef __attribute__((ext_vector_type(4))) int   v4i;

#define N_PTS 16384
#define M_PTS 16384
#define DIM   512
#define BM    256          // rows per block
#define BN    128          // cols per chunk
#define KCH   16           // K-chunk staged in LDS
#define LDST  20           // padded LDS row stride (floats): 16B aligned, conflict-free
#define NSPLIT 8           // column strips (grid.y)
#define NCHUNK (DIM / KCH) // 32

#if __has_builtin(__builtin_amdgcn_global_load_async_to_lds_b128)
#define USE_ASYNC 1
#else
#define USE_ASYNC 0
#endif

typedef __attribute__((address_space(1))) v4i g_v4i;   // global int4 (AS1)
typedef __attribute__((address_space(3))) v4i l_v4i;   // LDS int4 (AS3)

__device__ __forceinline__ g_v4i* glb_cast4(const float* p) {
  return (g_v4i*)(unsigned long long)p;
}
__device__ __forceinline__ l_v4i* lds_cast4(float* p) {
  // low 32 bits of a flat shared pointer are the LDS byte address
  return (l_v4i*)(unsigned)(unsigned long long)p;
}

// order-preserving float <-> uint key so integer atomicMin == float min
__device__ __forceinline__ unsigned f32_key(float f) {
  unsigned u = __float_as_uint(f);
  return u ^ (unsigned)(((int)u >> 31) | 0x80000000);
}
__device__ __forceinline__ float key_f32(unsigned k) {
  unsigned u = (k & 0x80000000u) ? (k ^ 0x80000000u) : ~k;
  return __uint_as_float(u);
}

// ---------------- Kernel A: row sums of squares + init per-row min keys ----------------
__global__ __launch_bounds__(256)
void kc_sumsq_init(const float* __restrict__ x, const float* __restrict__ y,
                   unsigned* __restrict__ wmin, float* __restrict__ xsq,
                   float* __restrict__ ysq) {
  int gwave = (int)((blockIdx.x * blockDim.x + threadIdx.x) >> 5);
  int lane  = threadIdx.x & 31;
  if (gwave >= N_PTS + M_PTS) return;
  const float* src = (gwave < N_PTS) ? (x + (size_t)gwave * DIM)
                                     : (y + (size_t)(gwave - N_PTS) * DIM);
  float s = 0.f;
  #pragma unroll 4
  for (int k = lane; k < DIM; k += 32) { float v = src[k]; s = fmaf(v, v, s); }
  #pragma unroll
  for (int off = 16; off; off >>= 1) s += __shfl_xor(s, off, 32);
  if (lane == 0) {
    if (gwave < N_PTS) { xsq[gwave] = s; wmin[gwave] = 0xFFFFFFFFu; }
    else               { ysq[gwave - N_PTS] = s; }
  }
}

// ---------------- Kernel B: fused WMMA GEMM + per-row min, double-buffered LDS ----------------
__global__ __launch_bounds__(256)
void kc_gemm_min(const float* __restrict__ x, const float* __restrict__ y,
                 const float* __restrict__ ysq, unsigned* __restrict__ wmin) {
  __shared__ float xs[2][BM * LDST];    // 2 x 20 KB
  __shared__ float ys_s[2][BN * LDST];  // 2 x 10 KB
  __shared__ unsigned smin[BM];

  const int tid  = threadIdx.x;
  const int lane = tid & 31;
  const int wave = tid >> 5;
  const int wm   = wave >> 1;   // wave row (0..3): 64 rows each
  const int wn   = wave & 1;    // wave col (0..1): 64 cols each
  const int half = lane >> 4;
  const int l16  = lane & 15;

  const int rowbase = blockIdx.x * BM;
  const int strip   = blockIdx.y * (M_PTS / NSPLIT);

  // cooperative loader mapping: 4 threads per row, one float4 each
  const int ldrow = tid >> 2;          // 0..63
  const int ldq   = (tid & 3) * 4;     // float4 column within K-chunk

  float rmin[4][8];
  #pragma unroll
  for (int mi = 0; mi < 4; ++mi)
    #pragma unroll
    for (int r = 0; r < 8; ++r) rmin[mi][r] = __builtin_inff();

#if !USE_ASYNC
  v4f rx[4], ry[2];  // register staging for fallback path
#endif

  for (int cc = 0; cc < (M_PTS / NSPLIT) / BN; ++cc) {
    const int colbase = strip + cc * BN;

    v8f acc[4][4];
    #pragma unroll
    for (int mi = 0; mi < 4; ++mi)
      #pragma unroll
      for (int ni = 0; ni < 4; ++ni)
        #pragma unroll
        for (int r = 0; r < 8; ++r) acc[mi][ni][r] = 0.f;

#if USE_ASYNC
    // prologue: async-fill buffer 0 with chunk 0 (ASYNCcnt-tracked, direct to LDS)
    {
      #pragma unroll
      for (int i = 0; i < 4; ++i)
        __builtin_amdgcn_global_load_async_to_lds_b128(
            glb_cast4(x + (size_t)(rowbase + ldrow + i * 64) * DIM + ldq),
            lds_cast4(&xs[0][(ldrow + i * 64) * LDST + ldq]), 0, 0);
      #pragma unroll
      for (int i = 0; i < 2; ++i)
        __builtin_amdgcn_global_load_async_to_lds_b128(
            glb_cast4(y + (size_t)(colbase + ldrow + i * 64) * DIM + ldq),
            lds_cast4(&ys_s[0][(ldrow + i * 64) * LDST + ldq]), 0, 0);
    }
#else
    #pragma unroll
    for (int i = 0; i < 4; ++i)
      rx[i] = *(const v4f*)(x + (size_t)(rowbase + ldrow + i * 64) * DIM + ldq);
    #pragma unroll
    for (int i = 0; i < 2; ++i)
      ry[i] = *(const v4f*)(y + (size_t)(colbase + ldrow + i * 64) * DIM + ldq);
#endif

    for (int ch = 0; ch < NCHUNK; ++ch) {
      const int p = ch & 1;
#if USE_ASYNC
#if __has_builtin(__builtin_amdgcn_s_wait_asynccnt)
      __builtin_amdgcn_s_wait_asynccnt(0);
#else
      asm volatile("s_wait_asynccnt 0x0" ::: "memory");
#endif
      __syncthreads();  // chunk ch resident in buf p; all waves done reading buf p^1
      if (ch + 1 < NCHUNK) {
        const int kb = (ch + 1) * KCH;
        #pragma unroll
        for (int i = 0; i < 4; ++i)
          __builtin_amdgcn_global_load_async_to_lds_b128(
              glb_cast4(x + (size_t)(rowbase + ldrow + i * 64) * DIM + kb + ldq),
              lds_cast4(&xs[p ^ 1][(ldrow + i * 64) * LDST + ldq]), 0, 0);
        #pragma unroll
        for (int i = 0; i < 2; ++i)
          __builtin_amdgcn_global_load_async_to_lds_b128(
              glb_cast4(y + (size_t)(colbase + ldrow + i * 64) * DIM + kb + ldq),
              lds_cast4(&ys_s[p ^ 1][(ldrow + i * 64) * LDST + ldq]), 0, 0);
      }
#else
      // fallback: regs hold chunk ch -> store to buf p, prefetch ch+1 into regs
      #pragma unroll
      for (int i = 0; i < 4; ++i)
        *(v4f*)(&xs[p][(ldrow + i * 64) * LDST + ldq]) = rx[i];
      #pragma unroll
      for (int i = 0; i < 2; ++i)
        *(v4f*)(&ys_s[p][(ldrow + i * 64) * LDST + ldq]) = ry[i];
      if (ch + 1 < NCHUNK) {
        const int kb = (ch + 1) * KCH;
        #pragma unroll
        for (int i = 0; i < 4; ++i)
          rx[i] = *(const v4f*)(x + (size_t)(rowbase + ldrow + i * 64) * DIM + kb + ldq);
        #pragma unroll
        for (int i = 0; i < 2; ++i)
          ry[i] = *(const v4f*)(y + (size_t)(colbase + ldrow + i * 64) * DIM + kb + ldq);
      }
      __syncthreads();
#endif
      // compute chunk ch from buf p: 4 kk-steps x 16 WMMA
      #pragma unroll
      for (int kk = 0; kk < KCH; kk += 4) {
        v2f a[4], b[4];
        #pragma unroll
        for (int mi = 0; mi < 4; ++mi)
          a[mi] = *(const v2f*)(&xs[p][(wm * 64 + mi * 16 + l16) * LDST + kk + 2 * half]);
        #pragma unroll
        for (int ni = 0; ni < 4; ++ni)
          b[ni] = *(const v2f*)(&ys_s[p][(wn * 64 + ni * 16 + l16) * LDST + kk + 2 * half]);
        #pragma unroll
        for (int mi = 0; mi < 4; ++mi)
          #pragma unroll
          for (int ni = 0; ni < 4; ++ni)
            acc[mi][ni] = __builtin_amdgcn_wmma_f32_16x16x4_f32(
                false, a[mi], false, b[ni], (short)0, acc[mi][ni], false, false);
      }
    }

    // epilogue: t = ||y_j||^2 - 2 x.y ; min over this 128-col chunk per row
    #pragma unroll
    for (int mi = 0; mi < 4; ++mi) {
      float m[8];
      #pragma unroll
      for (int ni = 0; ni < 4; ++ni) {
        float q = ysq[colbase + wn * 64 + ni * 16 + l16];
        #pragma unroll
        for (int r = 0; r < 8; ++r) {
          float v = fmaf(-2.f, acc[mi][ni][r], q);
          m[r] = ni ? fminf(m[r], v) : v;
        }
      }
      #pragma unroll
      for (int r = 0; r < 8; ++r) {
        float v = m[r];
        v = fminf(v, __shfl_xor(v, 1, 32));
        v = fminf(v, __shfl_xor(v, 2, 32));
        v = fminf(v, __shfl_xor(v, 4, 32));
        v = fminf(v, __shfl_xor(v, 8, 32));
        rmin[mi][r] = fminf(rmin[mi][r], v);
      }
    }
  }

  // block-level per-row combine (2 waves share each row), one global atomic per row
  if (tid < BM) smin[tid] = 0xFFFFFFFFu;
  __syncthreads();
  if (l16 == 0) {
    #pragma unroll
    for (int mi = 0; mi < 4; ++mi)
      #pragma unroll
      for (int r = 0; r < 8; ++r) {
        int rl = wm * 64 + mi * 16 + half * 8 + r;
        atomicMin(&smin[rl], f32_key(rmin[mi][r]));
      }
  }
  __syncthreads();
  if (tid < BM) atomicMin(&wmin[rowbase + tid], smin[tid]);
}

// ---------------- Kernel C: sqrt(min + ||x||^2), argmax with first-index ties ----------------
__global__ __launch_bounds__(1024)
void kc_reduce(const unsigned* __restrict__ wmin, const float* __restrict__ xsq,
               float* __restrict__ out) {
  __shared__ float sv[1024];
  __shared__ int   si[1024];
  int tid = threadIdx.x;
  float bv = -__builtin_inff();
  int   bi = 0x7FFFFFFF;
  for (int i = tid; i < N_PTS; i += 1024) {
    float sq = key_f32(wmin[i]) + xsq[i];
    float d  = __builtin_sqrtf(fmaxf(sq, 0.f));
    if (d > bv || (d == bv && i < bi)) { bv = d; bi = i; }
  }
  sv[tid] = bv; si[tid] = bi;
  __syncthreads();
  for (int s = 512; s > 0; s >>= 1) {
    if (tid < s) {
      float ov = sv[tid + s]; int oi = si[tid + s];
      if (ov > sv[tid] || (ov == sv[tid] && oi < si[tid])) { sv[tid] = ov; si[tid] = oi; }
    }
    __syncthreads();
  }
  if (tid == 0) { out[0] = sv[0]; out[1] = (float)si[0]; }
}

extern "C" void kernel_launch(void* const* d_in, const int* in_sizes, int n_in,
                              void* d_out, int out_size, void* d_ws, size_t ws_size,
                              hipStream_t stream) {
  const float* x = (const float*)d_in[0];
  const float* y = (const float*)d_in[1];
  unsigned* wmin = (unsigned*)d_ws;
  float* xsq = (float*)((char*)d_ws + (size_t)N_PTS * 4);
  float* ysq = (float*)((char*)d_ws + (size_t)N_PTS * 8);
  float* out = (float*)d_out;

  kc_sumsq_init<<<dim3((N_PTS + M_PTS) / 8), dim3(256), 0, stream>>>(x, y, wmin, xsq, ysq);
  kc_gemm_min<<<dim3(N_PTS / BM, NSPLIT), dim3(256), 0, stream>>>(x, y, ysq, wmin);
  kc_reduce<<<dim3(1), dim3(1024), 0, stream>>>(wmin, xsq, out);
}